// TDGAT_67662914781636
// MI455X (gfx1250) — compile-verified
//
#include <hip/hip_runtime.h>
#include <float.h>

typedef __attribute__((ext_vector_type(2))) float v2f;
typedef __attribute__((ext_vector_type(8))) float v8f;

#define FEAT 64
#define TPB  256

// ---------------------------------------------------------------- utilities
__global__ __launch_bounds__(TPB) void fill_kernel(float* __restrict__ p, int n, float v) {
  int i = blockIdx.x * TPB + threadIdx.x;
  if (i < n) p[i] = v;
}

__device__ inline void atomicMaxF(float* addr, float val) {
  // float max via signed/unsigned integer ordering trick; init must be -FLT_MAX
  if (val >= 0.0f) atomicMax((int*)addr, __float_as_int(val));
  else             atomicMin((unsigned int*)addr, __float_as_uint(val));
}

// ---------------------------------------------------------------- xp = X @ W (fp32 WMMA)
// One wave computes a 16-row tile of xp: 4 column tiles x 16 K-steps of
// v_wmma_f32_16x16x4_f32. W (64x64 fp32, 16KB) is staged in LDS per block.
__global__ __launch_bounds__(TPB) void gat_gemm_wmma(
    const float* __restrict__ X, const float* __restrict__ W,
    float* __restrict__ XP, int n_nodes)
{
  __shared__ float sW[FEAT * FEAT];
  for (int i = threadIdx.x; i < FEAT * FEAT; i += TPB) sW[i] = W[i];
  __syncthreads();

  const int wave = threadIdx.x >> 5;
  const int lane = threadIdx.x & 31;
  const int tile = blockIdx.x * (TPB / 32) + wave;
  const int row0 = tile * 16;
  if (row0 >= n_nodes) return;

  const int half = lane >> 4;   // 0: K pair {0,1}; 1: K pair {2,3} of each group of 4
  const int l16  = lane & 15;
  const int arow = row0 + l16;
  const bool av  = (arow < n_nodes);

  // A fragments: 16 k-steps, lane holds X[arow, 4*kk + 2*half + {0,1}]
  v2f a[16];
  const float* xr = X + (size_t)arow * FEAT + half * 2;
  #pragma unroll
  for (int kk = 0; kk < 16; ++kk) {
    v2f t = {0.0f, 0.0f};
    if (av) t = *(const v2f*)(xr + kk * 4);
    a[kk] = t;
  }

  #pragma unroll
  for (int ct = 0; ct < 4; ++ct) {           // 4 column tiles of 16
    const int n0 = ct * 16;
    v8f c = {};
    #pragma unroll
    for (int kk = 0; kk < 16; ++kk) {
      const int k0 = kk * 4 + half * 2;
      v2f b;
      b.x = sW[(k0 + 0) * FEAT + n0 + l16];  // B(K,N): row-major W in LDS
      b.y = sW[(k0 + 1) * FEAT + n0 + l16];
      c = __builtin_amdgcn_wmma_f32_16x16x4_f32(
              /*neg_a=*/false, a[kk], /*neg_b=*/false, b,
              /*c_mod=*/(short)0, c, /*reuse_a=*/false, /*reuse_b=*/false);
    }
    // C/D layout: VGPR i holds (M = i + 8*half, N = l16)
    #pragma unroll
    for (int i = 0; i < 8; ++i) {
      const int r = row0 + half * 8 + i;
      if (r < n_nodes) XP[(size_t)r * FEAT + n0 + l16] = c[i];
    }
  }
}

// ---------------------------------------------------------------- per-node alpha dots
__global__ __launch_bounds__(TPB) void gat_alpha(
    const float* __restrict__ XP, const float* __restrict__ a_src,
    const float* __restrict__ a_dst, float* __restrict__ as_n,
    float* __restrict__ ad_n, int n_nodes)
{
  const int w    = (blockIdx.x * TPB + threadIdx.x) >> 5;  // one wave per node
  const int lane = threadIdx.x & 31;
  if (w >= n_nodes) return;
  v2f v = *(const v2f*)(XP + (size_t)w * FEAT + lane * 2);
  float s = v.x * a_src[lane * 2] + v.y * a_src[lane * 2 + 1];
  float d = v.x * a_dst[lane * 2] + v.y * a_dst[lane * 2 + 1];
  #pragma unroll
  for (int off = 16; off > 0; off >>= 1) {
    s += __shfl_xor(s, off);
    d += __shfl_xor(d, off);
  }
  if (lane == 0) { as_n[w] = s; ad_n[w] = d; }
}

// ---------------------------------------------------------------- edge phase
__global__ __launch_bounds__(TPB) void gat_edge_max(
    const int* __restrict__ ei, int E, int n_nodes,
    const float* __restrict__ as_n, const float* __restrict__ ad_n,
    float* __restrict__ e_buf, float* __restrict__ m)
{
  const int i  = blockIdx.x * TPB + threadIdx.x;
  const int ET = E + n_nodes;
  if (i >= ET) return;
  int s, d;
  if (i < E) { s = ei[i]; d = ei[E + i]; } else { s = d = i - E; }  // self-loops appended
  float e = as_n[s] + ad_n[d];
  e = (e > 0.0f) ? e : 0.2f * e;            // leaky_relu, slope 0.2
  e_buf[i] = e;
  atomicMaxF(m + d, e);
}

__global__ __launch_bounds__(TPB) void gat_edge_expsum(
    const int* __restrict__ ei, int E, int n_nodes,
    const float* __restrict__ m, float* __restrict__ e_buf,
    float* __restrict__ denom)
{
  const int i  = blockIdx.x * TPB + threadIdx.x;
  const int ET = E + n_nodes;
  if (i >= ET) return;
  const int d = (i < E) ? ei[E + i] : (i - E);
  const float ex = __expf(e_buf[i] - m[d]);
  e_buf[i] = ex;
  atomicAdd(denom + d, ex);
}

// one wave per edge: gather xp[src] (L2-resident), scatter alpha*xp into out[dst]
__global__ __launch_bounds__(TPB) void gat_edge_scatter(
    const int* __restrict__ ei, int E, int n_nodes,
    const float* __restrict__ e_buf, const float* __restrict__ denom,
    const float* __restrict__ XP, float* __restrict__ OUT)
{
  const int w    = (blockIdx.x * TPB + threadIdx.x) >> 5;
  const int lane = threadIdx.x & 31;
  const int ET   = E + n_nodes;
  if (w >= ET) return;
  int s, d;
  if (w < E) { s = ei[w]; d = ei[E + w]; } else { s = d = w - E; }
  const float alpha = e_buf[w] / denom[d];
  v2f xv = *(const v2f*)(XP + (size_t)s * FEAT + lane * 2);
  float* o = OUT + (size_t)d * FEAT + lane * 2;
  atomicAdd(o,     alpha * xv.x);
  atomicAdd(o + 1, alpha * xv.y);
}

__global__ __launch_bounds__(TPB) void gat_bias_relu(
    float* __restrict__ h, const float* __restrict__ b, int n_elems)
{
  const int i = blockIdx.x * TPB + threadIdx.x;
  if (i >= n_elems) return;
  const float v = h[i] + b[i & (FEAT - 1)];
  h[i] = v > 0.0f ? v : 0.0f;
}

// ---------------------------------------------------------------- graph mean pool
__global__ __launch_bounds__(TPB) void gat_pool_sum(
    const float* __restrict__ h, const int* __restrict__ batch,
    float* __restrict__ out, float* __restrict__ cnt, int n_nodes)
{
  const int i = blockIdx.x * TPB + threadIdx.x;
  if (i >= n_nodes * FEAT) return;
  const int n = i >> 6, f = i & (FEAT - 1);
  const int g = batch[n];
  atomicAdd(out + (size_t)g * FEAT + f, h[i]);
  if (f == 0) atomicAdd(cnt + g, 1.0f);
}

__global__ __launch_bounds__(TPB) void gat_pool_div(
    float* __restrict__ out, const float* __restrict__ cnt, int n)
{
  const int i = blockIdx.x * TPB + threadIdx.x;
  if (i < n) out[i] /= fmaxf(cnt[i >> 6], 1.0f);
}

// ---------------------------------------------------------------- launcher
extern "C" void kernel_launch(void* const* d_in, const int* in_sizes, int n_in,
                              void* d_out, int out_size, void* d_ws, size_t ws_size,
                              hipStream_t stream)
{
  (void)n_in; (void)ws_size;
  const float* x     = (const float*)d_in[0];
  const int*   ei    = (const int*)d_in[1];
  const int*   batch = (const int*)d_in[2];
  const float* W1  = (const float*)d_in[3];
  const float* as1 = (const float*)d_in[4];
  const float* ad1 = (const float*)d_in[5];
  const float* b1  = (const float*)d_in[6];
  const float* W2  = (const float*)d_in[7];
  const float* as2 = (const float*)d_in[8];
  const float* ad2 = (const float*)d_in[9];
  const float* b2  = (const float*)d_in[10];

  const int N  = in_sizes[0] / FEAT;   // 50000
  const int E  = in_sizes[1] / 2;      // 800000 (row0 = src, row1 = dst)
  const int ET = E + N;                // + self-loops

  float* ws   = (float*)d_ws;
  float* xp   = ws;  ws += (size_t)N * FEAT;   // xp of current layer
  float* h    = ws;  ws += (size_t)N * FEAT;   // layer output / next layer input
  float* as_n = ws;  ws += N;
  float* ad_n = ws;  ws += N;
  float* m    = ws;  ws += N;
  float* den  = ws;  ws += N;
  float* e    = ws;  ws += ET;
  float* cnt  = ws;  ws += 128;

  float* outp = (float*)d_out;

  auto cdiv = [](long a, long b) { return (int)((a + b - 1) / b); };
  const int gemm_blocks = cdiv(cdiv(N, 16), TPB / 32);

  const float* feat = x;
  for (int layer = 0; layer < 2; ++layer) {
    const float* W  = layer ? W2  : W1;
    const float* as = layer ? as2 : as1;
    const float* ad = layer ? ad2 : ad1;
    const float* b  = layer ? b2  : b1;

    fill_kernel<<<cdiv(N, TPB), TPB, 0, stream>>>(m, N, -FLT_MAX);
    fill_kernel<<<cdiv(N, TPB), TPB, 0, stream>>>(den, N, 0.0f);
    gat_gemm_wmma<<<gemm_blocks, TPB, 0, stream>>>(feat, W, xp, N);
    gat_alpha<<<cdiv((long)N * 32, TPB), TPB, 0, stream>>>(xp, as, ad, as_n, ad_n, N);
    gat_edge_max<<<cdiv(ET, TPB), TPB, 0, stream>>>(ei, E, N, as_n, ad_n, e, m);
    gat_edge_expsum<<<cdiv(ET, TPB), TPB, 0, stream>>>(ei, E, N, m, e, den);
    fill_kernel<<<cdiv((long)N * FEAT, TPB), TPB, 0, stream>>>(h, N * FEAT, 0.0f);
    gat_edge_scatter<<<cdiv((long)ET * 32, TPB), TPB, 0, stream>>>(ei, E, N, e, den, xp, h);
    gat_bias_relu<<<cdiv((long)N * FEAT, TPB), TPB, 0, stream>>>(h, b, N * FEAT);
    feat = h;
  }

  fill_kernel<<<cdiv(out_size, TPB), TPB, 0, stream>>>(outp, out_size, 0.0f);
  fill_kernel<<<1, TPB, 0, stream>>>(cnt, 128, 0.0f);
  gat_pool_sum<<<cdiv((long)N * FEAT, TPB), TPB, 0, stream>>>(h, batch, outp, cnt, N);
  gat_pool_div<<<cdiv(out_size, TPB), TPB, 0, stream>>>(outp, cnt, out_size);
}